// RiemannianLIFNode_25202868093200
// MI455X (gfx1250) — compile-verified
//
#include <hip/hip_runtime.h>

typedef __attribute__((ext_vector_type(2))) float v2f;
typedef __attribute__((ext_vector_type(8))) float v8f;
typedef __attribute__((ext_vector_type(4))) unsigned int u32x4;
typedef __attribute__((ext_vector_type(4))) int i32x4;
typedef __attribute__((ext_vector_type(8))) int i32x8;

#define T_STEPS 128
#define NB      8            // time chunks of 16
#define BT      16
#define NT      64           // neurons per workgroup
#define NE      (2048 * 64)  // total independent neurons (N*D)
#define HSTR    72           // LDS history row stride (pad 64 -> 72: stagger banks)
#define VTH     1.0f
#define H_ALPHA 0.22360679774997896f  // 0.05 ** 0.5
#define INV_TAU 0.5f                  // 1 / TAU

#if defined(__AMDGCN__)
#define HAS_TDM __has_builtin(__builtin_amdgcn_tensor_load_to_lds)
#else
#define HAS_TDM 0
#endif

// ---------------------------------------------------------------------------
// Fractional LIF (Grunwald-Letnikov) over 128 steps.
// One workgroup = 64 neurons for all 128 steps.
//  * Inter-chunk GL history: V_WMMA_F32_16X16X4_F32 (Toeplitz coeff tile x
//    V-history tile), 4 chained K=4 WMMAs per past chunk.
//  * x staging: Tensor Data Mover 2D tile load (16 rows x 256B, row stride
//    NE*4B) into LDS, issued by wave 0 at chunk start so it overlaps the WMMA
//    phase; consumed after s_wait_tensorcnt + workgroup barrier.
//  * Intra-chunk triangle + spike nonlinearity: sequential VALU, one thread
//    per neuron, recent V values register-rolled (fully unrolled).
// ---------------------------------------------------------------------------
__global__ __launch_bounds__(128) void lif_gl_kernel(const float* __restrict__ x,
                                                     float* __restrict__ spikes) {
  __shared__ float cc[T_STEPS + 8];       // GL coefficients c[0..128]
  __shared__ float hist[T_STEPS * HSTR];  // V history: hist[k-1][n] = V_k
  __shared__ float Rbuf[BT * NT];         // inter-chunk contribution tile
  __shared__ float xstage[BT * NT];       // TDM-staged x tile for this chunk

  const int tid  = threadIdx.x;
  const int wave = tid >> 5;
  const int lane = tid & 31;
  const int half = lane >> 4;   // 0: lanes 0-15 (K 0,1), 1: lanes 16-31 (K 2,3)
  const int lr   = lane & 15;
  const int base = blockIdx.x * NT;

  if (tid == 0) {  // c[j] = (1 - 1.5/j) * c[j-1]
    float c = 1.0f;
    cc[0] = c;
    for (int j = 1; j <= T_STEPS; ++j) { c *= (1.0f - 1.5f / (float)j); cc[j] = c; }
  }
  __syncthreads();

  float V = 0.0f;  // thread tid < NT owns neuron (base + tid)

  for (int b = 0; b < NB; ++b) {
    const int kb = b * BT;  // history-row base of this chunk

    // ---- TDM: stage this chunk's x tile (16 x 64 f32) into LDS ---------
#if HAS_TDM
    if (tid < 32) {  // wave 0 only (scalar branch; other waves jump around)
      const unsigned lds_off = (unsigned)(size_t)(void*)xstage;
      const unsigned long long ga =
          (unsigned long long)(const void*)(x + (size_t)kb * NE + base);
      // D# group0: count=1 (valid), lds_addr, global_addr[56:0], type=2
      u32x4 g0 = {1u, lds_off, (unsigned)ga,
                  ((unsigned)(ga >> 32) & 0x01FFFFFFu) | 0x80000000u};
      // D# group1: data_size=4B; tensor [NE x T], stride0=NE; tile 64 x 16
      i32x8 g1 = {(int)(2u << 16),                       // data_size=2 (4B)
                  (int)((NE & 0xFFFFu) << 16),           // tensor_dim0[15:0]
                  (int)((NE >> 16) | (T_STEPS << 16)),   // dim0[31:16], dim1[15:0]
                  (int)((unsigned)NT << 16),             // dim1[31:16]=0, tile_dim0
                  (int)BT,                               // tile_dim1=16, tile_dim2=0
                  (int)NE,                               // dim0_stride[31:0]
                  0, 0};                                 // dim0_stride[47:32], dim1_stride
      i32x4 z4 = {0, 0, 0, 0};
#if __has_include(<hip/amd_detail/amd_gfx1250_TDM.h>)
      i32x8 z8 = {0, 0, 0, 0, 0, 0, 0, 0};
      __builtin_amdgcn_tensor_load_to_lds(g0, g1, z4, z4, z8, 0);
#else
      __builtin_amdgcn_tensor_load_to_lds(g0, g1, z4, z4, 0);
#endif
    }
#else
    // Fallback: cooperative staging loads (with prefetch)
    if (tid < 32)
      __builtin_prefetch(&x[(size_t)(kb + (tid >> 1)) * NE + base + (tid & 1) * 32], 0, 1);
    for (int i = tid; i < BT * NT; i += 128)
      xstage[i] = x[(size_t)(kb + i / NT) * NE + base + (i % NT)];
#endif

    // ---- Phase A: inter-chunk history via f32 WMMA ---------------------
    // R[t, n] = sum_{bp<b} C_{b-bp} x Vhist[bp],  C_d[t][s] = cc[16d + t - s]
    v8f acc = {0.f, 0.f, 0.f, 0.f, 0.f, 0.f, 0.f, 0.f};
    const int col = wave * 16 + lr;  // this wave's 16 neuron columns
    for (int bp = 0; bp < b; ++bp) {           // uniform bound: EXEC all-1s
      const int d16  = (b - bp) * 16;
      const int srow = bp * BT;
#pragma unroll
      for (int kk = 0; kk < 4; ++kk) {         // K = 16 as 4 chained K=4 WMMAs
        const int k0 = kk * 4 + 2 * half;      // ISA 16x4 f32 A/B lane layout
        v2f a, bb;
        a.x  = cc[d16 + lr - k0];              // A: Toeplitz coeff tile
        a.y  = cc[d16 + lr - (k0 + 1)];
        bb.x = hist[(srow + k0) * HSTR + col]; // B: V-history tile
        bb.y = hist[(srow + k0 + 1) * HSTR + col];
        acc = __builtin_amdgcn_wmma_f32_16x16x4_f32(false, a, false, bb,
                                                    (short)0, acc, false, false);
      }
    }
#pragma unroll
    for (int r = 0; r < 8; ++r)                 // C/D layout: VGPR r -> M = r + 8*half
      Rbuf[(r + 8 * half) * NT + col] = acc[r];

#if HAS_TDM
    if (tid < 32) __builtin_amdgcn_s_wait_tensorcnt(0);  // x tile landed in LDS
#endif
    __syncthreads();

    // ---- Phase B: sequential intra-chunk update (1 thread per neuron) --
    if (tid < NT) {
      float vloc[BT];  // rolling recent V values, kept in registers (unrolled)
#pragma unroll
      for (int tl = 0; tl < BT; ++tl) {
        const int k = kb + tl + 1;  // 1-based step
        float right = Rbuf[tl * NT + tid];
#pragma unroll
        for (int j = 0; j < tl; ++j) right += cc[tl - j] * vloc[j];
        const float xk = xstage[tl * NT + tid];
        const float sp = (V > VTH) ? 1.0f : 0.0f;
        const float f  = INV_TAU * (-V + xk - VTH * sp);
        const float Vn = f * H_ALPHA - right;
        vloc[tl] = Vn;
        hist[(kb + tl) * HSTR + tid] = Vn;
        spikes[(size_t)(k - 1) * NE + base + tid] = (Vn > VTH) ? 1.0f : 0.0f;
        V = Vn;
      }
    }
    __syncthreads();  // next chunk's WMMA reads this chunk's hist / xstage
  }
}

// ---------------------------------------------------------------------------
// Poincare-ball exponential map: one wave32 per row (D=64 -> 2 elems/lane),
// row reductions via __shfl_xor. y2 and x.y collapse algebraically:
//   sc = tanh(lam*|v|/2)/|v|,  y = sc*v,  y2 = sc^2*Sum(v^2),  xy = sc*Sum(zv)
// ---------------------------------------------------------------------------
__global__ __launch_bounds__(256) void expmap_kernel(const float* __restrict__ v,
                                                     const float* __restrict__ z,
                                                     float* __restrict__ out,
                                                     int nrows) {
  const int row  = blockIdx.x * 8 + (threadIdx.x >> 5);
  const int lane = threadIdx.x & 31;
  if (row >= nrows) return;
  const float eps = 1e-7f;
  const float* zr = z + (size_t)row * 64;
  const float* vr = v + (size_t)row * 64;
  float z0 = zr[lane], z1 = zr[lane + 32];
  float w0 = vr[lane], w1 = vr[lane + 32];
  float s_z2 = z0 * z0 + z1 * z1;
  float s_v2 = w0 * w0 + w1 * w1;
  float s_zv = z0 * w0 + z1 * w1;
#pragma unroll
  for (int m = 16; m >= 1; m >>= 1) {
    s_z2 += __shfl_xor(s_z2, m, 32);
    s_v2 += __shfl_xor(s_v2, m, 32);
    s_zv += __shfl_xor(s_zv, m, 32);
  }
  const float x2  = s_z2;
  const float lam = 2.0f / fmaxf(1.0f - x2, eps);
  const float vn  = fmaxf(sqrtf(s_v2), eps);
  const float t   = tanhf(0.5f * lam * vn);
  const float sc  = t / vn;
  const float xy  = sc * s_zv;
  const float y2  = sc * sc * s_v2;
  const float nA  = 1.0f + 2.0f * xy + y2;
  const float nB  = (1.0f - x2) * sc;
  const float inv = 1.0f / fmaxf(1.0f + 2.0f * xy + x2 * y2, eps);
  float* o = out + (size_t)row * 64;
  o[lane]      = (nA * z0 + nB * w0) * inv;
  o[lane + 32] = (nA * z1 + nB * w1) * inv;
}

extern "C" void kernel_launch(void* const* d_in, const int* in_sizes, int n_in,
                              void* d_out, int out_size, void* d_ws, size_t ws_size,
                              hipStream_t stream) {
  (void)n_in; (void)out_size; (void)d_ws; (void)ws_size;
  const float* x_seq = (const float*)d_in[0];  // [128, 2048, 64] f32
  const float* v_seq = (const float*)d_in[1];  // [2048, 64] f32
  const float* z_seq = (const float*)d_in[2];  // [2048, 64] f32
  float* out    = (float*)d_out;
  float* spikes = out;                               // [128, 2048, 64]
  float* z_out  = out + (size_t)T_STEPS * NE;        // [2048, 64]

  const int nrows = in_sizes[1] / 64;                // 2048
  lif_gl_kernel<<<NE / NT, 128, 0, stream>>>(x_seq, spikes);
  expmap_kernel<<<(nrows + 7) / 8, 256, 0, stream>>>(v_seq, z_seq, z_out, nrows);
}